// Raymarcher_55353538510938
// MI455X (gfx1250) — compile-verified
//
#include <hip/hip_runtime.h>
#include <hip/hip_bf16.h>
#include <stdint.h>

#define BOUND      2.0f
#define GRIDN      128
#define GRID3      2097152u      // 128^3
#define MIN_NEAR   0.2f
#define M_STEPS    512
#define MAX_SAMP   1100000u
#define DT         0.003383f
#define N_RAYS     16384

typedef float vfloat4 __attribute__((ext_vector_type(4)));

__device__ __forceinline__ unsigned part1by2(unsigned x) {
    x &= 1023u;
    x = (x | (x << 16)) & 0x030000FFu;
    x = (x | (x << 8))  & 0x0300F00Fu;
    x = (x | (x << 4))  & 0x030C30C3u;
    x = (x | (x << 2))  & 0x09249249u;
    return x;
}

__device__ __forceinline__ float safe_comp(float d) {
    return (fabsf(d) < 1e-8f) ? 1e-8f : d;
}

// ---------------- kernel 0: zero the scatter target (NT stores) -------------
__global__ void __launch_bounds__(256) zero_xyzs(float* p, int n) {
    int i = blockIdx.x * blockDim.x + threadIdx.x;
    if (i < n) __builtin_nontemporal_store(0.0f, p + i);
}

// ---------------- kernel 1: march, flags via wave32 ballot -------------------
__global__ void __launch_bounds__(512)
march(const float* __restrict__ rays, const int* __restrict__ bitfield,
      float* __restrict__ t_out, float* __restrict__ dists_out,
      unsigned* __restrict__ ws_flags, unsigned* __restrict__ ws_counts,
      float* __restrict__ ws_near) {
    const int r = blockIdx.x;
    const int s = threadIdx.x;

    const float ox = rays[r * 6 + 0], oy = rays[r * 6 + 1], oz = rays[r * 6 + 2];
    const float dx = rays[r * 6 + 3], dy = rays[r * 6 + 4], dz = rays[r * 6 + 5];

    const float ix = 1.0f / safe_comp(dx);
    const float iy = 1.0f / safe_comp(dy);
    const float iz = 1.0f / safe_comp(dz);

    const float t0x = (-BOUND - ox) * ix, t1x = (BOUND - ox) * ix;
    const float t0y = (-BOUND - oy) * iy, t1y = (BOUND - oy) * iy;
    const float t0z = (-BOUND - oz) * iz, t1z = (BOUND - oz) * iz;

    float near = fmaxf(fmaxf(fminf(t0x, t1x), fmaxf(fminf(t0y, t1y), fminf(t0z, t1z))), MIN_NEAR);
    float far  = fmaxf(fminf(fmaxf(t0x, t1x), fminf(fmaxf(t0y, t1y), fmaxf(t0z, t1z))), near + 0.0001f);

    const float t = near + ((float)s + 0.5f) * DT;
    const float x = ox + t * dx;
    const float y = oy + t * dy;
    const float z = oz + t * dz;
    const float mx = fmaxf(fabsf(x), fmaxf(fabsf(y), fabsf(z)));

    // level = clip(ceil(log2(max(mx,1))), 0, 1): exactly (mx > 1) for CASCADE=2
    const int level = (mx > 1.0f) ? 1 : 0;
    const float mip = level ? 2.0f : 1.0f;

    const int gx = (int)fminf(fmaxf(floorf((x / mip * 0.5f + 0.5f) * (float)GRIDN), 0.0f), 127.0f);
    const int gy = (int)fminf(fmaxf(floorf((y / mip * 0.5f + 0.5f) * (float)GRIDN), 0.0f), 127.0f);
    const int gz = (int)fminf(fmaxf(floorf((z / mip * 0.5f + 0.5f) * (float)GRIDN), 0.0f), 127.0f);

    const unsigned idx = (unsigned)level * GRID3 +
        (part1by2((unsigned)gx) | (part1by2((unsigned)gy) << 1) | (part1by2((unsigned)gz) << 2));

    // 2 MB bitfield: L2-resident random gather (regular-temporal load, keep cached)
    const int word = bitfield[idx >> 3];
    const int occ = (word >> (idx & 7)) & 1;

    const bool flag = (t < far) && (mx <= BOUND) && (occ != 0);

    // streaming write-once outputs: non-temporal so they don't evict the
    // bitfield / flag working set from L2
    const size_t o = (size_t)r * M_STEPS + (size_t)s;
    __builtin_nontemporal_store(t, t_out + o);
    __builtin_nontemporal_store(DT, dists_out + o);

    // wave32 ballot: 16 waves/block -> 512-bit validity mask per ray
    const unsigned bal = __builtin_amdgcn_ballot_w32(flag);
    __shared__ unsigned warr[16];
    const int wave = s >> 5, lane = s & 31;
    if (lane == 0) warr[wave] = bal;
    __syncthreads();
    if (s == 0) {
        unsigned tot = 0;
        #pragma unroll
        for (int i = 0; i < 16; ++i) {
            ws_flags[(size_t)r * 16 + i] = warr[i];
            tot += __popc(warr[i]);
        }
        ws_counts[r] = tot;
        ws_near[r] = near;
    }
}

// ---------------- kernel 2: global scan over 16384 ray counts ---------------
__global__ void __launch_bounds__(1024)
scan(const unsigned* __restrict__ counts, unsigned* __restrict__ rayexcl,
     unsigned* __restrict__ wvalid, float* __restrict__ out_whole) {
    __shared__ unsigned sums[1024];
    const int tid = threadIdx.x;

    unsigned cnt[16], inc[16];
    unsigned run = 0;
    #pragma unroll
    for (int i = 0; i < 16; ++i) {
        cnt[i] = counts[tid * 16 + i];
        run += cnt[i];
        inc[i] = run;
    }
    sums[tid] = run;
    __syncthreads();

    // Hillis-Steele inclusive scan over 1024 thread totals (LDS)
    for (int off = 1; off < 1024; off <<= 1) {
        unsigned v = sums[tid];
        unsigned a = (tid >= off) ? sums[tid - off] : 0u;
        __syncthreads();
        sums[tid] = v + a;
        __syncthreads();
    }
    const unsigned chunk_excl = sums[tid] - run;

    #pragma unroll
    for (int i = 0; i < 16; ++i) {
        const int r = tid * 16 + i;
        const unsigned incl = chunk_excl + inc[i];
        const unsigned wv = (incl < MAX_SAMP) ? 1u : 0u;
        rayexcl[r] = incl - cnt[i];
        wvalid[r] = wv;
        out_whole[r] = wv ? 1.0f : 0.0f;
    }
}

// ---------------- kernel 3: compaction / scatter ----------------------------
__global__ void __launch_bounds__(512)
compact(const float* __restrict__ rays, const unsigned* __restrict__ ws_flags,
        const unsigned* __restrict__ rayexcl, const unsigned* __restrict__ wvalid,
        const float* __restrict__ ws_near, const float* __restrict__ focalp,
        float* __restrict__ xyzs, float* __restrict__ rayvalid_out) {
    const int r = blockIdx.x;
    const int s = threadIdx.x;

    __shared__ unsigned fw[16];
    if (s < 16) {
        // CDNA5 async global->LDS copy (ASYNCcnt-tracked), GVS addressing
        unsigned ldsa = (unsigned)(size_t)&fw[s];
        unsigned voff = (unsigned)s * 4u;
        unsigned long long gp = (unsigned long long)(uintptr_t)(ws_flags + (size_t)r * 16);
        asm volatile("global_load_async_to_lds_b32 %0, %1, %2 offset:0"
                     :: "v"(ldsa), "v"(voff), "s"(gp) : "memory");
        asm volatile("s_wait_asynccnt 0" ::: "memory");
    }
    __syncthreads();

    const int wave = s >> 5, lane = s & 31;
    const unsigned myw = fw[wave];
    const unsigned bit = (myw >> lane) & 1u;
    const unsigned valid = bit & wvalid[r];

    __builtin_nontemporal_store(valid ? 1.0f : 0.0f,
                                rayvalid_out + (size_t)r * M_STEPS + s);

    if (valid) {
        unsigned base = 0;
        for (int i = 0; i < wave; ++i) base += __popc(fw[i]);
        base += __popc(myw & ((1u << lane) - 1u));
        const unsigned pos = rayexcl[r] + base;

        const float near = ws_near[r];
        const float t = near + ((float)s + 0.5f) * DT;
        const float ox = rays[r * 6 + 0], oy = rays[r * 6 + 1], oz = rays[r * 6 + 2];
        const float dx = rays[r * 6 + 3], dy = rays[r * 6 + 4], dz = rays[r * 6 + 5];
        const float focal = focalp[0];

        vfloat4 v;
        v.x = ox + t * dx;
        v.y = oy + t * dy;
        v.z = oz + t * dz;
        v.w = t / focal;
        // 16B-aligned 128-bit non-temporal scatter
        __builtin_nontemporal_store(v, (vfloat4*)(xyzs) + pos);
    }
}

extern "C" void kernel_launch(void* const* d_in, const int* in_sizes, int n_in,
                              void* d_out, int out_size, void* d_ws, size_t ws_size,
                              hipStream_t stream) {
    const float* rays     = (const float*)d_in[0];
    const int*   bitfield = (const int*)d_in[1];
    const float* focal    = (const float*)d_in[2];

    float* out = (float*)d_out;
    const size_t RS = (size_t)N_RAYS * M_STEPS;          // 8388608
    float* out_xyzs     = out;                            // 4,400,000
    float* out_rayvalid = out + 4400000;                  // 8,388,608
    float* out_t        = out + 4400000 + RS;             // 8,388,608
    float* out_dists    = out + 4400000 + 2 * RS;         // 8,388,608
    float* out_whole    = out + 4400000 + 3 * RS;         // 16,384

    unsigned char* ws = (unsigned char*)d_ws;
    unsigned* ws_flags   = (unsigned*)(ws);                          // 16384*16*4 = 1 MiB
    unsigned* ws_counts  = (unsigned*)(ws + 1048576);                // 64 KiB
    float*    ws_near    = (float*)   (ws + 1048576 + 65536);        // 64 KiB
    unsigned* ws_rayexcl = (unsigned*)(ws + 1048576 + 131072);       // 64 KiB
    unsigned* ws_wvalid  = (unsigned*)(ws + 1048576 + 196608);       // 64 KiB

    zero_xyzs<<<(4400000 + 255) / 256, 256, 0, stream>>>(out_xyzs, 4400000);

    march<<<N_RAYS, 512, 0, stream>>>(rays, bitfield, out_t, out_dists,
                                      ws_flags, ws_counts, ws_near);

    scan<<<1, 1024, 0, stream>>>(ws_counts, ws_rayexcl, ws_wvalid, out_whole);

    compact<<<N_RAYS, 512, 0, stream>>>(rays, ws_flags, ws_rayexcl, ws_wvalid,
                                        ws_near, focal, out_xyzs, out_rayvalid);
}